// RangeLoss_21595095564576
// MI455X (gfx1250) — compile-verified
//
#include <hip/hip_runtime.h>

typedef __attribute__((ext_vector_type(2))) float v2f;
typedef __attribute__((ext_vector_type(8))) float v8f;

#define CR_HI 1.05f
#define CR_LO 0.95f

__device__ __forceinline__ float adjust01(float p, float t) {
    // ti = where(|pi|<0.01 & ti==0, pi, ti); then where(0.99<pi<1.01 & ti==1, pi, ti)
    if (fabsf(p) < 0.01f && t == 0.0f) t = p;
    if (p > 0.99f && p < 1.01f && t == 1.0f) t = p;
    return t;
}

__device__ __forceinline__ float row_loss(const float4 p, const float4 t) {
    float t0 = adjust01(p.x, t.x);
    float t1 = adjust01(p.y, t.y);
    // cond_hi = p2 > 0.9 ; cond_rng = (p1*1.05 > t1) & (p1*0.95 < t1)  [t1 already adjusted]
    bool cond = (p.z > 0.9f) || ((p.y * CR_HI > t1) && (p.y * CR_LO < t1));
    if (cond) t1 = p.y;
    float d0 = p.x - t0;
    float d1 = p.y - t1;
    float d2 = p.z - t.z;
    float d3 = p.w - t.w;
    return d0 * d0 + d1 * d1 + d2 * d2 + d3 * d3;
}

// Full wave32 sum via the matrix pipe (5x V_WMMA_F32_16X16X4_F32).
// Stage 1: A[m][0..1]=s_m (lanes 0-15), A[m][2..3]=s_{m+16} (lanes 16-31); B=0.5
//          => D1[m][n] = s_m + s_{m+16} = R[m], replicated over n.
// Stage 2: A=ones; B built from pairs of D1 VGPRs. Each WMMA j adds
//          R[2j] + R[2j+8] + R[2j+1] + R[2j+9]; j=0..3 covers all 16 R exactly once.
//          => c[*] = sum over all 32 lanes, replicated in every lane.
__device__ __forceinline__ float wave_reduce_wmma(float s) {
    v2f a;    a[0] = s;    a[1] = s;
    v2f half; half[0] = 0.5f; half[1] = 0.5f;
    v8f z = {};
    v8f r1 = __builtin_amdgcn_wmma_f32_16x16x4_f32(
        false, a, false, half, (short)0, z, false, false);

    v2f ones; ones[0] = 1.0f; ones[1] = 1.0f;
    v8f c = {};
    v2f b;
    b[0] = r1[0]; b[1] = r1[1];
    c = __builtin_amdgcn_wmma_f32_16x16x4_f32(false, ones, false, b, (short)0, c, false, false);
    b[0] = r1[2]; b[1] = r1[3];
    c = __builtin_amdgcn_wmma_f32_16x16x4_f32(false, ones, false, b, (short)0, c, false, false);
    b[0] = r1[4]; b[1] = r1[5];
    c = __builtin_amdgcn_wmma_f32_16x16x4_f32(false, ones, false, b, (short)0, c, false, false);
    b[0] = r1[6]; b[1] = r1[7];
    c = __builtin_amdgcn_wmma_f32_16x16x4_f32(false, ones, false, b, (short)0, c, false, false);
    return c[0];
}

__global__ __launch_bounds__(256) void range_loss_main(
    const float4* __restrict__ preds, const float4* __restrict__ target,
    double* __restrict__ blockPartials, long long rows) {
    const long long stride = (long long)gridDim.x * blockDim.x;
    long long i = (long long)blockIdx.x * blockDim.x + threadIdx.x;

    float acc = 0.0f;
    // 4-way unrolled streaming loop: 8 independent b128 loads in flight per iter.
    for (; i + 3 * stride < rows; i += 4 * stride) {
        float4 p0 = preds[i];
        float4 p1 = preds[i + stride];
        float4 p2 = preds[i + 2 * stride];
        float4 p3 = preds[i + 3 * stride];
        float4 q0 = target[i];
        float4 q1 = target[i + stride];
        float4 q2 = target[i + 2 * stride];
        float4 q3 = target[i + 3 * stride];
        acc += row_loss(p0, q0);
        acc += row_loss(p1, q1);
        acc += row_loss(p2, q2);
        acc += row_loss(p3, q3);
    }
    for (; i < rows; i += stride) acc += row_loss(preds[i], target[i]);

    // All 256 threads reach here (EXEC all-1s): WMMA wave reduction is legal.
    float wsum = wave_reduce_wmma(acc);

    __shared__ float waveSums[8];
    int lane = threadIdx.x & 31;
    int wave = threadIdx.x >> 5;
    if (lane == 0) waveSums[wave] = wsum;
    __syncthreads();
    if (threadIdx.x == 0) {
        double b = 0.0;
        #pragma unroll
        for (int k = 0; k < 8; ++k) b += (double)waveSums[k];
        blockPartials[blockIdx.x] = b;  // plain store; every slot rewritten each launch
    }
}

__global__ __launch_bounds__(256) void range_loss_finalize(
    const double* __restrict__ partials, int numPartials,
    float* __restrict__ out, double invCount) {
    __shared__ double sd[256];
    double s = 0.0;
    for (int i = threadIdx.x; i < numPartials; i += 256) s += partials[i];
    sd[threadIdx.x] = s;
    __syncthreads();
    for (int off = 128; off > 0; off >>= 1) {
        if ((int)threadIdx.x < off) sd[threadIdx.x] += sd[threadIdx.x + off];
        __syncthreads();
    }
    if (threadIdx.x == 0) out[0] = (float)(sd[0] * invCount);
}

extern "C" void kernel_launch(void* const* d_in, const int* in_sizes, int n_in,
                              void* d_out, int out_size, void* d_ws, size_t ws_size,
                              hipStream_t stream) {
    const float4* preds  = (const float4*)d_in[0];
    const float4* target = (const float4*)d_in[1];
    const long long nElems = (long long)in_sizes[0];  // 32,000,000
    const long long rows   = nElems / 4;              // 8,000,000

    int grid = 2048;  // 16 KB of fp64 partials in d_ws; ~15 rows per thread
    int maxBlocks = (int)(ws_size / sizeof(double));
    if (maxBlocks >= 1 && grid > maxBlocks) grid = maxBlocks;

    double* partials = (double*)d_ws;
    range_loss_main<<<grid, 256, 0, stream>>>(preds, target, partials, rows);
    range_loss_finalize<<<1, 256, 0, stream>>>(partials, grid, (float*)d_out,
                                               1.0 / (double)nElems);
}